// NCModel_8624294330467
// MI455X (gfx1250) — compile-verified
//
#include <hip/hip_runtime.h>
#include <hip/hip_bf16.h>

typedef __attribute__((ext_vector_type(2))) float v2f;
typedef __attribute__((ext_vector_type(8))) float v8f;

// ---------------------------------------------------------------------------
// GEMM + bias:  C[M,N] = A[M,K] @ W[K,N] + b[N]   (fp32, V_WMMA_F32_16X16X4_F32)
// One wave -> one 16x16 C tile. K stepped by 4 per WMMA.
// A-frag: lane holds row (lane&15), K offset (lane>>4)*2, 2 consecutive floats.
// B-frag: lane holds col (lane&15), same K offsets, row stride N.
// C/D   : VGPR v holds M = tile*16 + v + (lane>=16 ? 8 : 0), N = tile*16 + (lane&15).
// ---------------------------------------------------------------------------
template <int K>
__global__ __launch_bounds__(256) void gemm_bias_wmma(
    const float* __restrict__ A, const float* __restrict__ W,
    const float* __restrict__ bias, float* __restrict__ C, int M, int N) {
  const int lane  = threadIdx.x & 31;
  const int wave  = threadIdx.x >> 5;
  const int tile  = blockIdx.x * (blockDim.x >> 5) + wave;
  const int nT    = N >> 4;
  const int mTile = tile / nT;
  const int nTile = tile - mTile * nT;
  if (mTile * 16 >= M) return;

  const int row   = mTile * 16 + (lane & 15);   // A row for this lane
  const int col   = nTile * 16 + (lane & 15);   // B/C column for this lane
  const int kOff  = (lane >> 4) * 2;            // 0 or 2
  const float* __restrict__ Arow = A + (size_t)row * K;

  v8f acc = {};
  for (int k = 0; k < K; k += 4) {
    v2f a, b;
    a.x = Arow[k + kOff];
    a.y = Arow[k + kOff + 1];
    b.x = W[(size_t)(k + kOff) * N + col];
    b.y = W[(size_t)(k + kOff + 1) * N + col];
    acc = __builtin_amdgcn_wmma_f32_16x16x4_f32(
        /*neg_a=*/false, a, /*neg_b=*/false, b,
        /*c_mod=*/(short)0, acc, /*reuse_a=*/false, /*reuse_b=*/false);
  }

  const float bv      = bias[col];
  const int   rowBase = mTile * 16 + ((lane >> 4) << 3);  // +0 or +8
#pragma unroll
  for (int v = 0; v < 8; ++v)
    C[(size_t)(rowBase + v) * N + col] = acc[v] + bv;
}

// ---------------------------------------------------------------------------
// Edge scatter-add: one wave per edge, lanes stride the D-wide feature row.
// acc[dst[e], :] += feat[src[e], :]   via non-returning global_atomic_add_f32.
// Destination accumulators (20MB / 102MB) are L2-resident on MI455X (192MB L2).
// ---------------------------------------------------------------------------
template <int D>
__global__ __launch_bounds__(256) void edge_scatter(
    const float* __restrict__ feat, const int* __restrict__ src,
    const int* __restrict__ dst, float* __restrict__ acc, int nEdges) {
  const int lane = threadIdx.x & 31;
  const int edge = blockIdx.x * (blockDim.x >> 5) + (threadIdx.x >> 5);
  if (edge >= nEdges) return;
  const int s = src[edge];
  const int d = dst[edge];
  const float* __restrict__ fr = feat + (size_t)s * D;
  float* __restrict__ ar = acc + (size_t)d * D;
#pragma unroll
  for (int j = 0; j < D / 32; ++j)
    atomicAdd(&ar[lane + 32 * j], fr[lane + 32 * j]);
}

// In-degree counts (as float, to match the reference's float divide).
__global__ __launch_bounds__(256) void edge_count(
    const int* __restrict__ dst, float* __restrict__ cnt, int nEdges) {
  int e = blockIdx.x * blockDim.x + threadIdx.x;
  if (e < nEdges) atomicAdd(&cnt[dst[e]], 1.0f);
}

// mean (acc /= cnt, 0 where cnt==0) + optional leaky_relu(0.01), in place.
__global__ __launch_bounds__(256) void mean_act(
    float* __restrict__ acc, const float* __restrict__ cnt,
    long long total, int D, int doLrelu) {
  long long i = (long long)blockIdx.x * blockDim.x + threadIdx.x;
  if (i >= total) return;
  const float c = cnt[i / D];
  float v = (c > 0.0f) ? acc[i] / c : 0.0f;
  if (doLrelu) v = (v > 0.0f) ? v : 0.01f * v;
  acc[i] = v;
}

// ---------------------------------------------------------------------------
// Problem constants (match the reference).
// ---------------------------------------------------------------------------
static constexpr int   kNChem  = 200000;
static constexpr int   kNGene  = 20000;
static constexpr int   kNEdges = 1000000;
static constexpr int   kIn     = 128;
static constexpr int   kHid    = 256;
static constexpr int   kOut    = 128;

extern "C" void kernel_launch(void* const* d_in, const int* in_sizes, int n_in,
                              void* d_out, int out_size, void* d_ws, size_t ws_size,
                              hipStream_t stream) {
  (void)in_sizes; (void)n_in; (void)out_size; (void)ws_size;

  const int*   src_cg     = (const int*)  d_in[0];
  const int*   dst_cg     = (const int*)  d_in[1];
  const int*   src_gc     = (const int*)  d_in[2];
  const int*   dst_gc     = (const int*)  d_in[3];
  const float* embed_chem = (const float*)d_in[4];
  // d_in[5] embed_gene : dead (layer-1 gc branch discarded)
  const float* W1_cg      = (const float*)d_in[6];
  const float* b1_cg      = (const float*)d_in[7];
  // d_in[8..9]  W1_gc/b1_gc : dead
  // d_in[10..11] W2_cg/b2_cg : dead (layer-2 cg branch discarded)
  const float* W2_gc      = (const float*)d_in[12];
  const float* b2_gc      = (const float*)d_in[13];
  float*       out        = (float*)d_out;

  // Workspace layout (floats)
  float* ws       = (float*)d_ws;
  float* Wh1      = ws;                                  // [200000,256]
  float* geneAcc  = Wh1     + (size_t)kNChem * kHid;     // [20000,256]
  float* geneCnt  = geneAcc + (size_t)kNGene * kHid;     // [20000]
  float* Wh2      = geneCnt + kNGene;                    // [20000,128]
  float* chemCnt  = Wh2     + (size_t)kNGene * kOut;     // [200000]

  // Zero the accumulators / counts / output (graph-capturable memset nodes).
  hipMemsetAsync(geneAcc, 0, ((size_t)kNGene * kHid + kNGene) * sizeof(float), stream);
  hipMemsetAsync(chemCnt, 0, (size_t)kNChem * sizeof(float), stream);
  hipMemsetAsync(out,     0, (size_t)kNChem * kOut * sizeof(float), stream);

  // 1) Wh1 = embed_chem @ W1_cg + b1_cg      [200000,128]x[128,256]
  {
    const int tiles  = (kNChem / 16) * (kHid / 16);   // 12500*16 = 200000
    const int blocks = tiles / 8;                     // 8 waves/block
    gemm_bias_wmma<kIn><<<blocks, 256, 0, stream>>>(embed_chem, W1_cg, b1_cg,
                                                    Wh1, kNChem, kHid);
  }

  // 2) cg-edge mean into gene nodes
  edge_count<<<(kNEdges + 255) / 256, 256, 0, stream>>>(dst_cg, geneCnt, kNEdges);
  edge_scatter<kHid><<<kNEdges / 8, 256, 0, stream>>>(Wh1, src_cg, dst_cg,
                                                      geneAcc, kNEdges);

  // 3) h_gene = leaky_relu(mean)  (in place on geneAcc)
  {
    const long long total = (long long)kNGene * kHid;
    mean_act<<<(unsigned)((total + 255) / 256), 256, 0, stream>>>(
        geneAcc, geneCnt, total, kHid, /*doLrelu=*/1);
  }

  // 4) Wh2 = h_gene @ W2_gc + b2_gc          [20000,256]x[256,128]
  {
    const int tiles  = (kNGene / 16) * (kOut / 16);   // 1250*8 = 10000
    const int blocks = tiles / 8;
    gemm_bias_wmma<kHid><<<blocks, 256, 0, stream>>>(geneAcc, W2_gc, b2_gc,
                                                     Wh2, kNGene, kOut);
  }

  // 5) gc-edge mean into chem nodes -> output
  edge_count<<<(kNEdges + 255) / 256, 256, 0, stream>>>(dst_gc, chemCnt, kNEdges);
  edge_scatter<kOut><<<kNEdges / 8, 256, 0, stream>>>(Wh2, src_gc, dst_gc,
                                                      out, kNEdges);
  {
    const long long total = (long long)kNChem * kOut;
    mean_act<<<(unsigned)((total + 255) / 256), 256, 0, stream>>>(
        out, chemCnt, total, kOut, /*doLrelu=*/0);
  }
}